// TemporalMemoryWithMLP_58806692217158
// MI455X (gfx1250) — compile-verified
//
#include <hip/hip_runtime.h>
#include <math.h>

// Problem dims
#define N_B    8
#define C_DIM  512
#define M_DIM  2000
#define HW     4096           // 64*64
#define T_TOK  (N_B * HW)     // 32768
#define MT     (M_DIM / 16)   // 125 m-tiles
#define CT     (C_DIM / 16)   // 32 c-tiles
#define EPSV   1e-12f
#define SHRK   0.0025f

// Output packing (flat concat in return order)
#define OUT_ELEMS (N_B * C_DIM * HW)   // 16,777,216
#define ATT_ELEMS (N_B * M_DIM * HW)   // 65,536,000
#define OUT_OFF 0
#define ATT_OFF OUT_ELEMS
#define MEM_OFF (OUT_ELEMS + ATT_ELEMS)

typedef __attribute__((ext_vector_type(2))) float v2f;
typedef __attribute__((ext_vector_type(8))) float v8f;

// LDS layout (floats). Pitches chosen even (8B-aligned ds_load_b64 for WMMA A
// operands) and with bank-stride != 0 mod 64 to avoid LDS bank conflicts.
#define ATT_P 2002
#define XN_P  514
#define S_ATT   0
#define S_XN    (16 * ATT_P)                 // aliased as out tile in phase 5
#define S_RED   (S_XN + 16 * XN_P)
#define S_SCL   (S_RED + 16 * 8)
#define SMEM_FLOATS (S_SCL + 16)             // 40,400 floats = 161,600 B

__device__ __forceinline__ v8f wmma_f32(v2f a, v2f b, v8f c) {
  return __builtin_amdgcn_wmma_f32_16x16x4_f32(false, a, false, b,
                                               (short)0, c, false, false);
}

// ---------------- Kernel 1: wn = l2norm(weight, axis=1) ----------------
__global__ __launch_bounds__(256)
void normalize_weight_kernel(const float* __restrict__ w, float* __restrict__ wn) {
  __shared__ float red[256];
  const int r  = blockIdx.x;
  const int tx = threadIdx.x;
  const float* row = w + (size_t)r * C_DIM;
  float ss = 0.f;
  for (int c = tx; c < C_DIM; c += 256) { float v = row[c]; ss += v * v; }
  red[tx] = ss;
  __syncthreads();
  for (int s = 128; s > 0; s >>= 1) {
    if (tx < s) red[tx] += red[tx + s];
    __syncthreads();
  }
  const float inv = 1.f / fmaxf(sqrtf(red[0]), EPSV);
  for (int c = tx; c < C_DIM; c += 256) wn[(size_t)r * C_DIM + c] = row[c] * inv;
}

// ---------------- Kernel 2: mem_fea_align = wn @ wn^T ----------------
__global__ __launch_bounds__(32)
void mem_align_kernel(const float* __restrict__ wn, float* __restrict__ memo) {
  const int lane = threadIdx.x & 31;
  const int lo = lane & 15, hi = lane >> 4;
  const int a0 = blockIdx.y * 16, b0 = blockIdx.x * 16;
  const float* arow = wn + (size_t)(a0 + lo) * C_DIM + 2 * hi;
  const float* brow = wn + (size_t)(b0 + lo) * C_DIM + 2 * hi;
  v8f acc = {};
  for (int k = 0; k < C_DIM; k += 4) {
    v2f a = *(const v2f*)(arow + k);
    v2f b = *(const v2f*)(brow + k);
    acc = wmma_f32(a, b, acc);
  }
#pragma unroll
  for (int r = 0; r < 8; ++r)
    memo[(size_t)(a0 + r + 8 * hi) * M_DIM + b0 + lo] = acc[r];
}

// ------- Kernel 3: fused normalize(x) -> score -> softmax/shrink -> out -------
__global__ __launch_bounds__(128)
void fused_attention_kernel(const float* __restrict__ x, const float* __restrict__ wn,
                            float* __restrict__ out_nchw, float* __restrict__ att_nmhw) {
  extern __shared__ float smem[];
  float* att_s = smem + S_ATT;   // [16][ATT_P]
  float* xn_s  = smem + S_XN;    // [16][XN_P]; reused as out tile in phase 5
  float* red_s = smem + S_RED;   // [16][8]
  float* scl_s = smem + S_SCL;   // [16]

  const int tx   = threadIdx.x;
  const int wave = tx >> 5;
  const int lane = tx & 31;
  const int lo = lane & 15, hi = lane >> 4;

  const int t0  = blockIdx.x * 16;     // 16 consecutive tokens, same n (4096%16==0)
  const int n   = t0 / HW;
  const int hw0 = t0 % HW;
  const float* xblk = x + (size_t)n * C_DIM * HW + hw0;

  // Phase 1: load x tile (coalesced 64B runs), l2-normalize per token in LDS
  for (int idx = tx; idx < 16 * C_DIM; idx += 128) {
    const int i = idx & 15, c = idx >> 4;
    xn_s[i * XN_P + c] = xblk[(size_t)c * HW + i];
  }
  __syncthreads();
  {
    const int g = tx >> 3, j = tx & 7;
    float ss = 0.f;
    for (int c = j; c < C_DIM; c += 8) { float v = xn_s[g * XN_P + c]; ss += v * v; }
    red_s[g * 8 + j] = ss;
    __syncthreads();
    if (j == 0) {
      float s = 0.f;
#pragma unroll
      for (int k = 0; k < 8; ++k) s += red_s[g * 8 + k];
      scl_s[g] = 1.f / fmaxf(sqrtf(s), EPSV);
    }
    __syncthreads();
  }
  for (int idx = tx; idx < 16 * C_DIM; idx += 128) {
    const int i = idx & 15, c = idx >> 4;
    xn_s[i * XN_P + c] *= scl_s[i];
  }
  __syncthreads();

  // Phase 2: att_s[16][2000] = xn_tile @ wn^T  (f32 WMMA, K-step 4)
  for (int mt = wave; mt < MT; mt += 4) {
    const int m0 = mt * 16;
    const float* aptr = xn_s + lo * XN_P + 2 * hi;                  // A: xn rows
    const float* bptr = wn + (size_t)(m0 + lo) * C_DIM + 2 * hi;    // B: wn^T cols
    v8f acc = {};
    for (int k = 0; k < C_DIM; k += 4) {
      v2f a = *(const v2f*)(aptr + k);
      v2f b = *(const v2f*)(bptr + k);
      acc = wmma_f32(a, b, acc);
    }
#pragma unroll
    for (int r = 0; r < 8; ++r)
      att_s[(r + 8 * hi) * ATT_P + m0 + lo] = acc[r];
  }
  __syncthreads();

  // Phase 3: per-token softmax -> softshrink -> L1 sum (8 threads per token)
  {
    const int g = tx >> 3, j = tx & 7;
    float* row = att_s + g * ATT_P;
    float mx = -3.402823466e38f;
    for (int m = j; m < M_DIM; m += 8) mx = fmaxf(mx, row[m]);
    red_s[g * 8 + j] = mx;
    __syncthreads();
    if (j == 0) {
      float v = red_s[g * 8];
#pragma unroll
      for (int k = 1; k < 8; ++k) v = fmaxf(v, red_s[g * 8 + k]);
      scl_s[g] = v;
    }
    __syncthreads();
    mx = scl_s[g];
    float sum = 0.f;
    for (int m = j; m < M_DIM; m += 8) { float e = __expf(row[m] - mx); row[m] = e; sum += e; }
    __syncthreads();
    red_s[g * 8 + j] = sum;
    __syncthreads();
    if (j == 0) {
      float v = 0.f;
#pragma unroll
      for (int k = 0; k < 8; ++k) v += red_s[g * 8 + k];
      scl_s[g] = 1.f / v;
    }
    __syncthreads();
    const float invs = scl_s[g];
    float l1 = 0.f;
    for (int m = j; m < M_DIM; m += 8) {
      const float p = row[m] * invs;
      const float q = (p > SHRK) ? (p - SHRK) : ((p < -SHRK) ? (p + SHRK) : 0.f);
      row[m] = q;
      l1 += fabsf(q);
    }
    __syncthreads();
    red_s[g * 8 + j] = l1;
    __syncthreads();
    if (j == 0) {
      float v = 0.f;
#pragma unroll
      for (int k = 0; k < 8; ++k) v += red_s[g * 8 + k];
      scl_s[g] = 1.f / fmaxf(v, EPSV);
    }
    __syncthreads();
  }

  // Phase 4: apply L1 scale; stream att_spatial [N,M,H,W] (coalesced 64B runs)
  {
    float* attg = att_nmhw + (size_t)n * M_DIM * HW + hw0;
    for (int idx = tx; idx < 16 * M_DIM; idx += 128) {
      const int i = idx & 15, m = idx >> 4;
      const float v = att_s[i * ATT_P + m] * scl_s[i];
      att_s[i * ATT_P + m] = v;
      attg[(size_t)m * HW + i] = v;
    }
  }
  __syncthreads();

  // Phase 5: out[16][512] = att @ wn (K = 2000), then coalesced NCHW store
  float* out_s = xn_s;  // alias: xn no longer needed
  for (int ct = wave; ct < CT; ct += 4) {
    const int c0 = ct * 16;
    const float* aptr = att_s + lo * ATT_P + 2 * hi;            // A: att rows
    const float* bcol = wn + c0 + lo + (size_t)(2 * hi) * C_DIM; // B: wn columns
    v8f acc = {};
    for (int k = 0; k < M_DIM; k += 4) {
      v2f a = *(const v2f*)(aptr + k);
      v2f b;
      b.x = bcol[(size_t)k * C_DIM];
      b.y = bcol[(size_t)(k + 1) * C_DIM];
      acc = wmma_f32(a, b, acc);
    }
#pragma unroll
    for (int r = 0; r < 8; ++r)
      out_s[(r + 8 * hi) * XN_P + c0 + lo] = acc[r];
  }
  __syncthreads();
  {
    float* og = out_nchw + (size_t)n * C_DIM * HW + hw0;
    for (int idx = tx; idx < 16 * C_DIM; idx += 128) {
      const int i = idx & 15, c = idx >> 4;
      og[(size_t)c * HW + i] = out_s[i * XN_P + c];
    }
  }
}

extern "C" void kernel_launch(void* const* d_in, const int* in_sizes, int n_in,
                              void* d_out, int out_size, void* d_ws, size_t ws_size,
                              hipStream_t stream) {
  (void)in_sizes; (void)n_in; (void)out_size; (void)ws_size;
  const float* x = (const float*)d_in[0];       // [8,512,64,64]
  const float* w = (const float*)d_in[1];       // [2000,512]
  float* out = (float*)d_out;
  float* wn  = (float*)d_ws;                    // 2000*512*4 = 4 MB scratch

  normalize_weight_kernel<<<M_DIM, 256, 0, stream>>>(w, wn);
  mem_align_kernel<<<dim3(MT, MT), 32, 0, stream>>>(wn, out + MEM_OFF);
  const size_t smem_bytes = (size_t)SMEM_FLOATS * sizeof(float);
  fused_attention_kernel<<<T_TOK / 16, 128, smem_bytes, stream>>>(
      x, wn, out + OUT_OFF, out + ATT_OFF);
}